// TensorProduct_47648367182205
// MI455X (gfx1250) — compile-verified
//
#include <hip/hip_runtime.h>
#include <math.h>

typedef float v2f __attribute__((ext_vector_type(2)));
typedef float v8f __attribute__((ext_vector_type(8)));

#define WMMA_F32X4(a, b, c) \
    __builtin_amdgcn_wmma_f32_16x16x4_f32(false, (a), false, (b), (short)0, (c), false, false)

// ---------------------------------------------------------------------------
// Device-side construction of the real Clebsch-Gordan coupling matrices Ue/Uo
// (exact same math as the Python reference, in double precision).
// ws layout (floats): [0,256) U2e  [256,512) U2o  [512,1536) U4e  [1536,2560) U4o
// U2*: 9x9 padded to 16x16 row-major.  U4*: 25x25 padded to 32x32 row-major.
// ---------------------------------------------------------------------------
__device__ inline double dfact(int k) {
    double r = 1.0;
    for (int i = 2; i <= k; ++i) r *= (double)i;
    return r;
}

__device__ double cg_coeff(int l1, int m1, int l2, int m2, int L, int M) {
    if (m1 + m2 != M) return 0.0;
    double pref = sqrt((double)(2 * L + 1) * dfact(L + l1 - l2) * dfact(L - l1 + l2) *
                       dfact(l1 + l2 - L) / dfact(l1 + l2 + L + 1));
    pref *= sqrt(dfact(L + M) * dfact(L - M) * dfact(l1 - m1) * dfact(l1 + m1) *
                 dfact(l2 - m2) * dfact(l2 + m2));
    double s = 0.0;
    for (int k = 0; k <= l1 + l2 - L; ++k) {
        int a1 = l1 + l2 - L - k, a2 = l1 - m1 - k, a3 = l2 + m2 - k;
        int a4 = L - l2 + m1 + k, a5 = L - l1 - m2 + k;
        if (a1 < 0 || a2 < 0 || a3 < 0 || a4 < 0 || a5 < 0) continue;
        double d = dfact(k) * dfact(a1) * dfact(a2) * dfact(a3) * dfact(a4) * dfact(a5);
        s += ((k & 1) ? -1.0 : 1.0) / d;
    }
    return pref * s;
}

// r2c matrix entry: row = complex-m index (0..2L), col = real-m index (0..2L)
__device__ void r2c_entry(int L, int row, int col, double* re, double* im) {
    const double isq2 = 0.70710678118654752440;
    int mr = col - L;  // real m
    int mc = row - L;  // complex m
    *re = 0.0; *im = 0.0;
    if (mr < 0) {
        double sgn = (mr & 1) ? -1.0 : 1.0;  // (-1)^mr == (-1)^|mr|
        if (mc == -mr)      *im = isq2 * sgn;
        else if (mc == mr)  *im = -isq2;
    } else if (mr == 0) {
        if (mc == 0) *re = 1.0;
    } else {
        double sgn = (mr & 1) ? -1.0 : 1.0;
        if (mc == mr)       *re = isq2 * sgn;
        else if (mc == -mr) *re = isq2;
    }
}

// t[a,b,c] of _real_cg(lam, lam, L)
__device__ float real_cg_entry(int lam, int L, int a, int b, int c) {
    double tre = 0.0, tim = 0.0;
    for (int m1 = -lam; m1 <= lam; ++m1) {
        for (int m2 = -lam; m2 <= lam; ++m2) {
            int M = m1 + m2;
            if (M < -L || M > L) continue;
            double cg = cg_coeff(lam, m1, lam, m2, L, M);
            if (cg == 0.0) continue;
            double r1re, r1im, r2re, r2im, rLre, rLim;
            r2c_entry(lam, lam + m1, a, &r1re, &r1im);
            r2c_entry(lam, lam + m2, b, &r2re, &r2im);
            r2c_entry(L, L + M, c, &rLre, &rLim);
            rLim = -rLim;  // conj
            double pre = r1re * r2re - r1im * r2im;
            double pim = r1re * r2im + r1im * r2re;
            tre += (pre * rLre - pim * rLim) * cg;
            tim += (pre * rLim + pim * rLre) * cg;
        }
    }
    // parity of (l1+l2+L) = parity of L (since l1+l2 = 2*lam even)
    return (float)((L & 1) ? tim : tre);
}

__global__ __launch_bounds__(256) void build_u_kernel(float* __restrict__ ws) {
    float* u2e = ws;
    float* u2o = ws + 256;
    float* u4e = ws + 512;
    float* u4o = ws + 1536;
    const int tid = threadIdx.x;
    for (int i = tid; i < 2560; i += blockDim.x) ws[i] = 0.0f;
    __syncthreads();
    for (int e = tid; e < 81 + 625; e += blockDim.x) {
        int lam, P, PPAD, idx;
        float *ue, *uo;
        if (e < 81) { lam = 1; P = 9;  PPAD = 16; idx = e;      ue = u2e; uo = u2o; }
        else        { lam = 2; P = 25; PPAD = 32; idx = e - 81; ue = u4e; uo = u4o; }
        int I = idx / P, col = idx % P;
        int L;
        if (col < 1) L = 0; else if (col < 4) L = 1; else if (col < 9) L = 2;
        else if (col < 16) L = 3; else L = 4;
        int cM = col - L * L;
        int a = I / (2 * lam + 1), b = I % (2 * lam + 1);
        float v = real_cg_entry(lam, L, a, b, cM);
        if ((L & 1) == 0) ue[I * PPAD + col] = v;
        else              uo[I * PPAD + col] = v;
    }
}

// ---------------------------------------------------------------------------
// Group 0: out0[:, 0, 0:128] = f1_l0[:,0,128:256] * f2_l0[:,0,128:256]
// ---------------------------------------------------------------------------
__global__ __launch_bounds__(256) void tp_g0_kernel(const float* __restrict__ a,
                                                    const float* __restrict__ b,
                                                    float* __restrict__ out, int n) {
    int idx = blockIdx.x * blockDim.x + threadIdx.x;
    if (idx >= n * 128) return;
    int sample = idx >> 7, c = idx & 127;
    size_t src = (size_t)sample * 256 + 128 + c;
    out[(size_t)sample * 256 + c] = a[src] * b[src];
}

// ---------------------------------------------------------------------------
// Groups 1..3.  One 128-thread block (4 waves) handles 64 (sample,channel)
// columns; each wave owns 16 columns = WMMA N dimension.
// ---------------------------------------------------------------------------
// stacked-row J -> source feature element (rows beyond the group's real blocks are 0)
template <int G>
__device__ inline float ldX(int J, int sample, int ch,
                            const float* __restrict__ l0, const float* __restrict__ l1,
                            const float* __restrict__ l2, const float* __restrict__ l3) {
    if (J < 1) return l0[(size_t)sample * 256 + ch];
    if (J < 4) return l1[((size_t)sample * 3 + (J - 1)) * 128 + ch];
    if (G >= 2 && J < 9)  return l2[((size_t)sample * 5 + (J - 4)) * 64 + ch];
    if (G >= 3 && J < 16) return l3[((size_t)sample * 7 + (J - 9)) * 32 + ch];
    return 0.0f;
}

// channel offset of group G inside output level L's concatenation
template <int G>
__device__ inline int choffset(int L) {
    if (G == 1) return (L == 0) ? 128 : 0;
    if (G == 2) return (L == 0) ? 192 : ((L == 1) ? 64 : 0);
    return (L == 0) ? 224 : ((L == 1) ? 96 : ((L == 2) ? 32 : 0));
}

template <int G, int P, int PPAD, int Q, int W, int LO>
__global__ __launch_bounds__(128) void tp_group_kernel(
    const float* __restrict__ f1_0, const float* __restrict__ f1_1,
    const float* __restrict__ f1_2, const float* __restrict__ f1_3,
    const float* __restrict__ f2_0, const float* __restrict__ f2_1,
    const float* __restrict__ f2_2, const float* __restrict__ f2_3,
    const float* __restrict__ Ue, const float* __restrict__ Uo,
    float* __restrict__ out, int n) {
    constexpr int COLS = 64;
    constexpr int KB = PPAD / 4;            // k-blocks per 16-wide matmul
    constexpr int IMAX = (G + 1) * (G + 1); // output rows actually kept (L <= G)

    __shared__ float sUe[PPAD * PPAD], sUo[PPAD * PPAD];
    __shared__ float sE1[PPAD * COLS], sO1[PPAD * COLS];
    __shared__ float sE2[PPAD * COLS], sO2[PPAD * COLS];
    __shared__ float sCE[PPAD * COLS], sCO[PPAD * COLS];

    const int tid = threadIdx.x;
    for (int i = tid; i < PPAD * PPAD; i += 128) { sUe[i] = Ue[i]; sUo[i] = Uo[i]; }
    __syncthreads();

    const int wave = tid >> 5, lane = tid & 31;
    const int laneN = lane & 15, laneH = lane >> 4;
    const int bc = wave * 16 + laneN;                    // block column 0..63
    const int sample = blockIdx.x * (COLS / W) + bc / W;
    const int c = bc % W;
    const int ch = LO + c;

    // ---- Phase 1: uncouple  e/o = Ue/Uo * X   (WMMA f32 16x16x4) ----
    for (int s = 0; s < 2; ++s) {
        const float* l0 = s ? f2_0 : f1_0;
        const float* l1 = s ? f2_1 : f1_1;
        const float* l2 = s ? f2_2 : f1_2;
        const float* l3 = s ? f2_3 : f1_3;
        float* dE = s ? sE2 : sE1;
        float* dO = s ? sO2 : sO1;

        v2f bf[KB];
#pragma unroll
        for (int kb = 0; kb < KB; ++kb) {
            int j0 = kb * 4 + laneH * 2;
            bf[kb].x = ldX<G>(j0, sample, ch, l0, l1, l2, l3);
            bf[kb].y = ldX<G>(j0 + 1, sample, ch, l0, l1, l2, l3);
        }
#pragma unroll
        for (int rb = 0; rb < PPAD / 16; ++rb) {
            v8f aE = {};
            v8f aO = {};
            const int M = rb * 16 + laneN;
#pragma unroll
            for (int kb = 0; kb < KB; ++kb) {
                const int k0 = kb * 4 + laneH * 2;
                v2f ae; ae.x = sUe[M * PPAD + k0]; ae.y = sUe[M * PPAD + k0 + 1];
                aE = WMMA_F32X4(ae, bf[kb], aE);
                v2f ao; ao.x = sUo[M * PPAD + k0]; ao.y = sUo[M * PPAD + k0 + 1];
                aO = WMMA_F32X4(ao, bf[kb], aO);
            }
#pragma unroll
            for (int r = 0; r < 8; ++r) {
                dE[(rb * 16 + laneH * 8 + r) * COLS + bc] = aE[r];
                dO[(rb * 16 + laneH * 8 + r) * COLS + bc] = aO[r];
            }
        }
    }
    __syncthreads();

    // ---- Phase 2: per-column QxQ products (VALU). thread pair per column:
    //      half 0 -> ce = e1*e2 + o1*o2 ; half 1 -> co = e1*o2 + o1*e2
    {
        const int col = tid >> 1, half = tid & 1;
        const float* RA = half ? sO2 : sE2;  // multiplies e1
        const float* RB = half ? sE2 : sO2;  // multiplies o1
        float Ma[Q][Q], Mb[Q][Q];
#pragma unroll
        for (int i = 0; i < Q; ++i)
#pragma unroll
            for (int j = 0; j < Q; ++j) {
                Ma[i][j] = RA[(i * Q + j) * COLS + col];
                Mb[i][j] = RB[(i * Q + j) * COLS + col];
            }
        float* dst = half ? sCO : sCE;
#pragma unroll
        for (int i = 0; i < Q; ++i) {
            float e1r[Q], o1r[Q];
#pragma unroll
            for (int j = 0; j < Q; ++j) {
                e1r[j] = sE1[(i * Q + j) * COLS + col];
                o1r[j] = sO1[(i * Q + j) * COLS + col];
            }
#pragma unroll
            for (int k = 0; k < Q; ++k) {
                float acc = 0.0f;
#pragma unroll
                for (int j = 0; j < Q; ++j) acc += e1r[j] * Ma[j][k] + o1r[j] * Mb[j][k];
                dst[(i * Q + k) * COLS + col] = acc;
            }
        }
        // zero padded rows so no stale LDS data (NaN) enters the couple WMMA
        for (int i = tid; i < (PPAD - P) * COLS; i += 128) {
            sCE[P * COLS + i] = 0.0f;
            sCO[P * COLS + i] = 0.0f;
        }
    }
    __syncthreads();

    // ---- Phase 3: couple  fe/fo = Ue^T*ce / Uo^T*co, keep rows with L<=G ----
    {
        v2f bE[KB], bO[KB];
#pragma unroll
        for (int kb = 0; kb < KB; ++kb) {
            const int k0 = kb * 4 + laneH * 2;
            bE[kb].x = sCE[k0 * COLS + bc]; bE[kb].y = sCE[(k0 + 1) * COLS + bc];
            bO[kb].x = sCO[k0 * COLS + bc]; bO[kb].y = sCO[(k0 + 1) * COLS + bc];
        }
        v8f aFE = {};
        v8f aFO = {};
        const int M = laneN;  // only output rows 0..15 are ever kept (L <= 3)
#pragma unroll
        for (int kb = 0; kb < KB; ++kb) {
            const int k0 = kb * 4 + laneH * 2;
            v2f ae; ae.x = sUe[k0 * PPAD + M]; ae.y = sUe[(k0 + 1) * PPAD + M];  // Ue^T
            aFE = WMMA_F32X4(ae, bE[kb], aFE);
            v2f ao; ao.x = sUo[k0 * PPAD + M]; ao.y = sUo[(k0 + 1) * PPAD + M];  // Uo^T
            aFO = WMMA_F32X4(ao, bO[kb], aFO);
        }
#pragma unroll
        for (int r = 0; r < 8; ++r) {
            const int I = laneH * 8 + r;
            if (I < IMAX) {
                int L;
                if (I < 1) L = 0; else if (I < 4) L = 1; else if (I < 9) L = 2; else L = 3;
                const int mm = I - L * L;
                const float v = (L & 1) ? aFO[r] : aFE[r];
                const size_t base = (L == 0) ? 0
                                  : (L == 1) ? (size_t)n * 256
                                  : (L == 2) ? (size_t)n * 640
                                             : (size_t)n * 960;
                const int CHL = 256 >> L;
                out[base + ((size_t)sample * (2 * L + 1) + mm) * CHL + choffset<G>(L) + c] = v;
            }
        }
    }
}

// ---------------------------------------------------------------------------
extern "C" void kernel_launch(void* const* d_in, const int* in_sizes, int n_in,
                              void* d_out, int out_size, void* d_ws, size_t ws_size,
                              hipStream_t stream) {
    (void)n_in; (void)out_size; (void)ws_size;
    // setup_inputs order: f1_l0, f2_l0, f1_l1, f2_l1, f1_l2, f2_l2, f1_l3, f2_l3
    const float* f1_0 = (const float*)d_in[0];
    const float* f2_0 = (const float*)d_in[1];
    const float* f1_1 = (const float*)d_in[2];
    const float* f2_1 = (const float*)d_in[3];
    const float* f1_2 = (const float*)d_in[4];
    const float* f2_2 = (const float*)d_in[5];
    const float* f1_3 = (const float*)d_in[6];
    const float* f2_3 = (const float*)d_in[7];
    float* out = (float*)d_out;
    float* ws = (float*)d_ws;
    const int n = in_sizes[0] / 256;

    build_u_kernel<<<dim3(1), dim3(256), 0, stream>>>(ws);

    const float* u2e = ws;
    const float* u2o = ws + 256;
    const float* u4e = ws + 512;
    const float* u4o = ws + 1536;

    tp_g0_kernel<<<dim3((n * 128 + 255) / 256), dim3(256), 0, stream>>>(f1_0, f2_0, out, n);

    // <G, P, PPAD, Q, W, LO>; 64 columns per block
    tp_group_kernel<1, 9, 16, 3, 64, 64>
        <<<dim3((unsigned)((size_t)n * 64 / 64)), dim3(128), 0, stream>>>(
            f1_0, f1_1, f1_2, f1_3, f2_0, f2_1, f2_2, f2_3, u2e, u2o, out, n);
    tp_group_kernel<2, 9, 16, 3, 32, 32>
        <<<dim3((unsigned)((size_t)n * 32 / 64)), dim3(128), 0, stream>>>(
            f1_0, f1_1, f1_2, f1_3, f2_0, f2_1, f2_2, f2_3, u2e, u2o, out, n);
    tp_group_kernel<3, 25, 32, 5, 32, 0>
        <<<dim3((unsigned)((size_t)n * 32 / 64)), dim3(128), 0, stream>>>(
            f1_0, f1_1, f1_2, f1_3, f2_0, f2_1, f2_2, f2_3, u4e, u4o, out, n);
}